// CCBiMambaBlock_78314433675555
// MI455X (gfx1250) — compile-verified
//
#include <hip/hip_runtime.h>
#include <math.h>

// ---------------- problem constants (match reference) ----------------
#define BB    2
#define LL    1024
#define DM    512      // d_model
#define DI_   1024     // expand * d_model
#define NST   16       // d_state
#define DTR   32       // dt_rank
#define NCL   8        // n_clusters
#define MR    (BB*LL)  // 2048 rows

typedef __attribute__((ext_vector_type(2))) float        v2f;
typedef __attribute__((ext_vector_type(8))) float        v8f;
typedef __attribute__((ext_vector_type(4))) unsigned int v4u;
typedef __attribute__((ext_vector_type(4))) int          v4i;
typedef __attribute__((ext_vector_type(8))) int          v8i;

// ---------------- device helpers ----------------
__device__ __forceinline__ float siluf(float x)     { return x / (1.0f + expf(-x)); }
__device__ __forceinline__ float softplusf(float x) { return (x > 20.0f) ? x : log1pf(expf(x)); }
__device__ __forceinline__ float geluf(float x)     { return 0.5f * x * (1.0f + erff(x * 0.70710678118654752f)); }

// block-wide sum for blockDim.x == 256 (8 waves of 32)
__device__ __forceinline__ float block_sum(float v, float* sbuf) {
    const int tid = threadIdx.x;
#pragma unroll
    for (int off = 16; off > 0; off >>= 1) v += __shfl_down(v, off, 32);
    __syncthreads();                 // protect sbuf reuse across calls
    if ((tid & 31) == 0) sbuf[tid >> 5] = v;
    __syncthreads();
    float r = (tid < 8) ? sbuf[tid] : 0.0f;
    if (tid < 32) {
#pragma unroll
        for (int off = 4; off > 0; off >>= 1) r += __shfl_down(r, off, 32);
    }
    if (tid == 0) sbuf[0] = r;
    __syncthreads();
    return sbuf[0];
}

// ---------------- LayerNorm over last dim (DM=512), optional seq-flipped copy ----------------
__global__ __launch_bounds__(256) void ln2_kernel(const float* __restrict__ x,
                                                  const float* __restrict__ g,
                                                  const float* __restrict__ b,
                                                  float* __restrict__ out,
                                                  float* __restrict__ out_rev) {
    __shared__ float sbuf[32];
    const int row = blockIdx.x;
    const int tid = threadIdx.x;
    const float* xr = x + (size_t)row * DM;
    const float v0 = xr[tid], v1 = xr[tid + 256];
    const float mean = block_sum(v0 + v1, sbuf) * (1.0f / DM);
    const float d0 = v0 - mean, d1 = v1 - mean;
    const float var = block_sum(d0 * d0 + d1 * d1, sbuf) * (1.0f / DM);
    const float rstd = rsqrtf(var + 1e-5f);
    const float o0 = d0 * rstd * g[tid]       + b[tid];
    const float o1 = d1 * rstd * g[tid + 256] + b[tid + 256];
    out[(size_t)row * DM + tid]       = o0;
    out[(size_t)row * DM + tid + 256] = o1;
    if (out_rev) {
        const int bb = row / LL, ll = row % LL;
        const size_t orow = (size_t)bb * LL + (LL - 1 - ll);
        out_rev[orow * DM + tid]       = o0;
        out_rev[orow * DM + tid + 256] = o1;
    }
}

// ===================================================================================
// GEMM v1 (direct from global): one wave per 16x16 tile. For small/odd shapes.
// C = act(A[M,K] @ W[N,K]^T + bias) (+resid). ACT: 0=none,1=softplus,2=gelu.
// ===================================================================================
template <int ACT>
__global__ __launch_bounds__(256) void gemm_wmma_kernel(const float* __restrict__ A, int lda,
                                                        const float* __restrict__ W, int ldb,
                                                        const float* __restrict__ bias,
                                                        const float* __restrict__ resid,
                                                        float* __restrict__ C, int ldc, int col_off,
                                                        int M, int N, int K, int lrev) {
    const int wave    = blockIdx.x * (blockDim.x >> 5) + (threadIdx.x >> 5);
    const int tiles_n = N >> 4;
    if (wave >= (M >> 4) * tiles_n) return;   // wave-uniform: EXEC stays all-ones for WMMA
    const int tm   = wave / tiles_n;
    const int tn   = wave - tm * tiles_n;
    const int lane = threadIdx.x & 31;
    const int half = lane >> 4;               // 0: K=0,1 / 1: K=2,3 (32-bit A/B layout)
    const int lid  = lane & 15;

    const float* Ap = A + (size_t)(tm * 16 + lid) * lda + 2 * half;
    const float* Wp = W + (size_t)(tn * 16 + lid) * ldb + 2 * half;

    v8f acc0 = {}, acc1 = {};
    for (int k = 0; k < K; k += 8) {
        v2f a0 = *(const v2f*)(Ap + k);
        v2f b0 = *(const v2f*)(Wp + k);
        v2f a1 = *(const v2f*)(Ap + k + 4);
        v2f b1 = *(const v2f*)(Wp + k + 4);
        acc0 = __builtin_amdgcn_wmma_f32_16x16x4_f32(false, a0, false, b0, (short)0, acc0, false, false);
        acc1 = __builtin_amdgcn_wmma_f32_16x16x4_f32(false, a1, false, b1, (short)0, acc1, false, false);
    }
    v8f acc = acc0 + acc1;

    const int col  = tn * 16 + lid;
    const float bv = bias ? bias[col] : 0.0f;
#pragma unroll
    for (int r = 0; r < 8; ++r) {
        const int row = tm * 16 + r + 8 * half;   // C layout: VGPR r, halves give M=r / M=r+8
        float v = acc[r] + bv;
        if (ACT == 1) v = softplusf(v);
        if (ACT == 2) v = geluf(v);
        int orow = row;
        if (lrev) { const int bb = row / lrev; const int ll = row - bb * lrev; orow = bb * lrev + (lrev - 1 - ll); }
        const size_t off = (size_t)orow * ldc + col_off + col;
        if (resid) v += resid[off];
        C[off] = v;
    }
}

// ===================================================================================
// GEMM v2 (TDM-staged, double-buffered): block = 8 waves computes a 32x256 macro-tile
// (2M x 4N waves, each wave a 16x64 tile = 4 WMMA chains sharing one A fragment).
// A K-chunks (32 rows x 64 floats) are DMA'd into LDS by the Tensor Data Mover with
// LDS padding (row stride 66 dwords -> conflict-free ds_load_b64 A-fragment reads).
// Chunk c+1's DMA is issued before waiting on chunk c (s_wait_tensorcnt 1) so the
// tensor DMA overlaps the WMMA compute. B loads are software-pipelined one k-step
// ahead to relax s_wait_loadcnt pressure. Requires M%32==0, N%256==0, K%64==0.
// ===================================================================================
#define KC  64
#define LDP 66   // padded LDS row stride in floats (64 + 2, via TDM pad feature)

template <int ACT>
__global__ __launch_bounds__(256) void gemm_wmma_tdm_kernel(const float* __restrict__ A, int lda,
                                                            const float* __restrict__ W, int ldb,
                                                            const float* __restrict__ bias,
                                                            const float* __restrict__ resid,
                                                            float* __restrict__ C, int ldc, int col_off,
                                                            int M, int N, int K, int lrev) {
    __shared__ float sA[2][32 * LDP];         // 2 x 8.25 KB staged A chunks (double buffer)

    const int nblk = N >> 8;                  // N / 256
    const int bm   = blockIdx.x / nblk;
    const int bn   = blockIdx.x - bm * nblk;
    const int wave = threadIdx.x >> 5;
    const int lane = threadIdx.x & 31;
    const int half = lane >> 4, lid = lane & 15;
    const int wm   = wave >> 2;               // 0..1  (M sub-tile)
    const int wn   = wave & 3;                // 0..3  (N sub-tile)

    const int row0  = bm * 32;                // first A row of the macro tile
    const int mrow  = wm * 16 + lid;          // this lane's A row within macro tile
    const int ncol0 = bn * 256 + wn * 64;     // wave's first output column

    const float* Wp[4];
#pragma unroll
    for (int s = 0; s < 4; ++s) Wp[s] = W + (size_t)(ncol0 + s * 16 + lid) * ldb + 2 * half;

    // ---- TDM descriptor group 1 (invariant across chunks) ----
    // data_size=4B (code 2), pad_enable, pad_interval=64 dwords (code 5), pad_amount=2 dwords (code 1)
    const unsigned td0 = (unsigned)K;                 // tensor_dim0 (elements, for OOB)
    const unsigned td1 = (unsigned)M;                 // tensor_dim1 (rows)
    const unsigned long long st0 = (unsigned long long)(unsigned)lda;  // tensor_dim0_stride
    v8i g1;
    g1[0] = (int)((2u << 16) | (1u << 20) | (5u << 22) | (1u << 25));
    g1[1] = (int)((td0 & 0xFFFFu) << 16);             // atomic_barrier_addr=0 | tensor_dim0[15:0]
    g1[2] = (int)((td0 >> 16) | ((td1 & 0xFFFFu) << 16));
    g1[3] = (int)((td1 >> 16) | ((unsigned)KC << 16));        // tensor_dim1[31:16] | tile_dim0
    g1[4] = (int)32;                                  // tile_dim1=32, tile_dim2=0
    g1[5] = (int)(st0 & 0xFFFFFFFFull);
    g1[6] = (int)((st0 >> 32) & 0xFFFFull);           // stride0[47:32] | stride1[15:0]=0
    g1[7] = 0;
    const v4i gz4 = {0, 0, 0, 0};                     // groups 2/3 unused (2D tensor)
    const v8i gz8 = {0, 0, 0, 0, 0, 0, 0, 0};         // unused trailing group (6-arg builtin)

    const unsigned lds_base = (unsigned)(size_t)(&sA[0][0]);  // low 32 bits of flat LDS addr

    // issue TDM for K-chunk c into LDS buffer `buf` (wave-level; EXEC ignored by TDM)
    auto issue_chunk = [&](int c, int buf) {
        const unsigned long long ga =
            (unsigned long long)(size_t)(A + (size_t)row0 * lda + c * KC);
        v4u g0;
        g0[0] = 1u;                                   // count=1 user descriptor
        g0[1] = lds_base + (unsigned)buf * (32u * LDP * 4u);
        g0[2] = (unsigned)(ga & 0xFFFFFFFFull);
        g0[3] = (unsigned)((ga >> 32) & 0x1FFFFFFu) | (2u << 30);  // global_addr[56:32] | type=2
        __builtin_amdgcn_tensor_load_to_lds(g0, g1, gz4, gz4, gz8, 0);
    };

    const int nchunk = K / KC;
    if (wave == 0) issue_chunk(0, 0);                 // prologue: chunk 0 in flight

    v8f acc[4] = {};
    for (int c = 0; c < nchunk; ++c) {
        const int k0 = c * KC;
        if (wave == 0) {
            if (c + 1 < nchunk) {
                issue_chunk(c + 1, (c + 1) & 1);      // next chunk's DMA overlaps compute
                __builtin_amdgcn_s_wait_tensorcnt(1); // chunk c done (in-order), c+1 in flight
            } else {
                __builtin_amdgcn_s_wait_tensorcnt(0); // last chunk: drain
            }
        }
        __syncthreads();                              // chunk c visible to all waves

        const float* sAr = &sA[c & 1][mrow * LDP + 2 * half];
        v2f bcur[4], bnxt[4];
#pragma unroll
        for (int s = 0; s < 4; ++s) { bcur[s] = *(const v2f*)(Wp[s] + k0); bnxt[s] = bcur[s]; }
#pragma unroll
        for (int k = 0; k < KC; k += 4) {
            v2f a = *(const v2f*)(sAr + k);           // ds_load_b64, conflict-free (padded rows)
            if (k + 4 < KC) {
#pragma unroll
                for (int s = 0; s < 4; ++s) bnxt[s] = *(const v2f*)(Wp[s] + k0 + k + 4);
            }
#pragma unroll
            for (int s = 0; s < 4; ++s)
                acc[s] = __builtin_amdgcn_wmma_f32_16x16x4_f32(false, a, false, bcur[s], (short)0, acc[s], false, false);
#pragma unroll
            for (int s = 0; s < 4; ++s) bcur[s] = bnxt[s];
        }
        __syncthreads();                              // buffer c&1 free for next TDM issue
    }

    // ---- epilogue ----
    const int grow0 = row0 + wm * 16;
#pragma unroll
    for (int s = 0; s < 4; ++s) {
        const int col  = ncol0 + s * 16 + lid;
        const float bv = bias ? bias[col] : 0.0f;
#pragma unroll
        for (int r = 0; r < 8; ++r) {
            const int row = grow0 + r + 8 * half;
            float v = acc[s][r] + bv;
            if (ACT == 1) v = softplusf(v);
            if (ACT == 2) v = geluf(v);
            int orow = row;
            if (lrev) { const int b2 = row / lrev; const int l2 = row - b2 * lrev; orow = b2 * lrev + (lrev - 1 - l2); }
            const size_t off = (size_t)orow * ldc + col_off + col;
            if (resid) v += resid[off];
            C[off] = v;
        }
    }
}

// ---------------- depthwise causal conv1d (K=4) + bias + silu ----------------
__global__ __launch_bounds__(256) void conv_silu_kernel(const float* __restrict__ xz,
                                                        const float* __restrict__ cw,
                                                        const float* __restrict__ cb,
                                                        float* __restrict__ xc) {
    const int idx = blockIdx.x * blockDim.x + threadIdx.x;
    if (idx >= MR * DI_) return;
    const int d   = idx & (DI_ - 1);
    const int row = idx >> 10;        // DI_ == 1024
    const int l   = row & (LL - 1);
    float acc = cb[d];
#pragma unroll
    for (int j = 0; j < 4; ++j) {
        const int li = l - 3 + j;
        if (li >= 0) acc += cw[d * 4 + j] * xz[(size_t)(row - 3 + j) * (2 * DI_) + d];
    }
    xc[idx] = siluf(acc);
}

// ---------------- selective-scan: one thread per (b, d) channel, 16-wide state in regs ----------------
__global__ __launch_bounds__(256) void ssm_scan_kernel(const float* __restrict__ delta,
                                                       const float* __restrict__ xc,
                                                       const float* __restrict__ xdbl,   // (MR,64): [dt|B|C]
                                                       const float* __restrict__ xz,     // (MR,2*DI_): z at DI_+d
                                                       const float* __restrict__ A_log,
                                                       const float* __restrict__ Dp,
                                                       float* __restrict__ y) {
    const int gid = blockIdx.x * blockDim.x + threadIdx.x;
    if (gid >= BB * DI_) return;
    const int b = gid >> 10;
    const int d = gid & (DI_ - 1);
    float Ac[NST], h[NST];
#pragma unroll
    for (int n = 0; n < NST; ++n) { Ac[n] = -expf(A_log[d * NST + n]); h[n] = 0.0f; }
    const float Dd = Dp[d];
    for (int l = 0; l < LL; ++l) {
        const size_t row = (size_t)b * LL + l;
        const float dl  = delta[row * DI_ + d];
        const float xcv = xc[row * DI_ + d];
        const float du  = dl * xcv;
        const float* Bm = xdbl + row * 64 + DTR;
        const float* Cm = Bm + NST;
        float yv = 0.0f;
#pragma unroll
        for (int n = 0; n < NST; ++n) {
            h[n] = expf(dl * Ac[n]) * h[n] + du * Bm[n];
            yv  += h[n] * Cm[n];
        }
        yv += xcv * Dd;
        const float zv = xz[row * (2 * DI_) + DI_ + d];
        y[row * DI_ + d] = yv * siluf(zv);
    }
}

// ---------------- cluster-center L2 norms (one block per center) ----------------
__global__ __launch_bounds__(256) void center_norm_kernel(const float* __restrict__ centers,
                                                          float* __restrict__ cnorm) {
    __shared__ float sbuf[32];
    const int c = blockIdx.x, tid = threadIdx.x;
    const float v0 = centers[c * DM + tid], v1 = centers[c * DM + tid + 256];
    const float s = block_sum(v0 * v0 + v1 * v1, sbuf);
    if (tid == 0) cnorm[c] = fmaxf(sqrtf(s), 1e-12f);
}

// ---------------- context clustering: cosine sim -> softmax(8) -> ctx -> LN(xn + a*ctx) ----------------
__global__ __launch_bounds__(256) void cc_kernel(const float* __restrict__ xn,
                                                 const float* __restrict__ proj,
                                                 const float* __restrict__ centers,
                                                 const float* __restrict__ cnorm,
                                                 const float* __restrict__ alpha_p,
                                                 const float* __restrict__ g,
                                                 const float* __restrict__ b,
                                                 float* __restrict__ cc_out) {
    __shared__ float sbuf[32];
    __shared__ float sw[NCL];
    const int row = blockIdx.x, tid = threadIdx.x;
    const float* pr = proj + (size_t)row * DM;
    const float p0 = pr[tid], p1 = pr[tid + 256];
    float pn = block_sum(p0 * p0 + p1 * p1, sbuf);
    pn = fmaxf(sqrtf(pn), 1e-12f);
    for (int c = 0; c < NCL; ++c) {
        const float* cp = centers + c * DM;
        const float dot = block_sum(p0 * cp[tid] + p1 * cp[tid + 256], sbuf);
        if (tid == 0) sw[c] = dot / (pn * cnorm[c]);
    }
    __syncthreads();
    if (tid == 0) {
        float m = sw[0];
        for (int c = 1; c < NCL; ++c) m = fmaxf(m, sw[c]);
        float ssum = 0.0f;
        for (int c = 0; c < NCL; ++c) { sw[c] = expf(sw[c] - m); ssum += sw[c]; }
        const float inv = 1.0f / ssum;
        for (int c = 0; c < NCL; ++c) sw[c] *= inv;
    }
    __syncthreads();
    const float alpha = alpha_p[0];
    float ctx0 = 0.0f, ctx1 = 0.0f;
#pragma unroll
    for (int c = 0; c < NCL; ++c) {
        ctx0 += sw[c] * centers[c * DM + tid];
        ctx1 += sw[c] * centers[c * DM + tid + 256];
    }
    const float t0 = xn[(size_t)row * DM + tid]       + alpha * ctx0;
    const float t1 = xn[(size_t)row * DM + tid + 256] + alpha * ctx1;
    const float mean = block_sum(t0 + t1, sbuf) * (1.0f / DM);
    const float d0 = t0 - mean, d1 = t1 - mean;
    const float var = block_sum(d0 * d0 + d1 * d1, sbuf) * (1.0f / DM);
    const float rstd = rsqrtf(var + 1e-5f);
    cc_out[(size_t)row * DM + tid]       = d0 * rstd * g[tid]       + b[tid];
    cc_out[(size_t)row * DM + tid + 256] = d1 * rstd * g[tid + 256] + b[tid + 256];
}

// ---------------- gated fusion + residual: x1 = x + g0*mamba_out + g1*cc_out ----------------
__global__ __launch_bounds__(256) void gate_fuse_kernel(const float* __restrict__ x,
                                                        const float* __restrict__ xn,
                                                        const float* __restrict__ gate_w,
                                                        const float* __restrict__ gate_b,
                                                        const float* __restrict__ mamba_out,
                                                        const float* __restrict__ cc_out,
                                                        float* __restrict__ x1) {
    __shared__ float sbuf[32];
    const int row = blockIdx.x, tid = threadIdx.x;
    const float* xr = xn + (size_t)row * DM;
    const float v0 = xr[tid], v1 = xr[tid + 256];
    float s0 = block_sum(v0 * gate_w[tid]      + v1 * gate_w[tid + 256],      sbuf);
    float s1 = block_sum(v0 * gate_w[DM + tid] + v1 * gate_w[DM + tid + 256], sbuf);
    s0 += gate_b[0]; s1 += gate_b[1];
    const float m  = fmaxf(s0, s1);
    const float e0 = expf(s0 - m), e1 = expf(s1 - m);
    const float g0 = e0 / (e0 + e1), g1 = e1 / (e0 + e1);
    const size_t o = (size_t)row * DM;
    x1[o + tid]       = x[o + tid]       + g0 * mamba_out[o + tid]       + g1 * cc_out[o + tid];
    x1[o + tid + 256] = x[o + tid + 256] + g0 * mamba_out[o + tid + 256] + g1 * cc_out[o + tid + 256];
}

// ---------------- host side ----------------
static inline void launch_gemm(int act,
                               const float* A, int lda, const float* W, int ldb,
                               const float* bias, const float* resid,
                               float* C, int ldc, int col_off,
                               int M, int N, int K, int lrev, hipStream_t s) {
    const bool v2 = (M % 32 == 0) && (N % 256 == 0) && (K % KC == 0);
    if (v2) {
        const int blocks = (M / 32) * (N / 256);
        if (act == 1)
            gemm_wmma_tdm_kernel<1><<<blocks, 256, 0, s>>>(A, lda, W, ldb, bias, resid, C, ldc, col_off, M, N, K, lrev);
        else if (act == 2)
            gemm_wmma_tdm_kernel<2><<<blocks, 256, 0, s>>>(A, lda, W, ldb, bias, resid, C, ldc, col_off, M, N, K, lrev);
        else
            gemm_wmma_tdm_kernel<0><<<blocks, 256, 0, s>>>(A, lda, W, ldb, bias, resid, C, ldc, col_off, M, N, K, lrev);
    } else {
        const int tiles  = (M / 16) * (N / 16);
        const int blocks = (tiles + 7) / 8;
        if (act == 1)
            gemm_wmma_kernel<1><<<blocks, 256, 0, s>>>(A, lda, W, ldb, bias, resid, C, ldc, col_off, M, N, K, lrev);
        else if (act == 2)
            gemm_wmma_kernel<2><<<blocks, 256, 0, s>>>(A, lda, W, ldb, bias, resid, C, ldc, col_off, M, N, K, lrev);
        else
            gemm_wmma_kernel<0><<<blocks, 256, 0, s>>>(A, lda, W, ldb, bias, resid, C, ldc, col_off, M, N, K, lrev);
    }
}

extern "C" void kernel_launch(void* const* d_in, const int* in_sizes, int n_in,
                              void* d_out, int out_size, void* d_ws, size_t ws_size,
                              hipStream_t stream) {
    (void)in_sizes; (void)n_in; (void)out_size; (void)ws_size;
    // inputs in setup_inputs() dict order
    const float* x         = (const float*)d_in[0];
    const float* norm1_g   = (const float*)d_in[1];
    const float* norm1_b   = (const float*)d_in[2];
    const float* fusion_w  = (const float*)d_in[3];
    const float* fusion_b  = (const float*)d_in[4];
    const float* cc_cent   = (const float*)d_in[5];
    const float* cc_pw     = (const float*)d_in[6];
    const float* cc_pb     = (const float*)d_in[7];
    const float* cc_ng     = (const float*)d_in[8];
    const float* cc_nb     = (const float*)d_in[9];
    const float* cc_alpha  = (const float*)d_in[10];
    const float* gate_w    = (const float*)d_in[11];
    const float* gate_b    = (const float*)d_in[12];
    const float* ffn_ng    = (const float*)d_in[13];
    const float* ffn_nb    = (const float*)d_in[14];
    const float* ffn_w1    = (const float*)d_in[15];
    const float* ffn_b1    = (const float*)d_in[16];
    const float* ffn_w2    = (const float*)d_in[17];
    const float* ffn_b2    = (const float*)d_in[18];
    const float* fm_in_w   = (const float*)d_in[19];
    const float* fm_conv_w = (const float*)d_in[20];
    const float* fm_conv_b = (const float*)d_in[21];
    const float* fm_xpw    = (const float*)d_in[22];
    const float* fm_dt_w   = (const float*)d_in[23];
    const float* fm_dt_b   = (const float*)d_in[24];
    const float* fm_A_log  = (const float*)d_in[25];
    const float* fm_D      = (const float*)d_in[26];
    const float* fm_out_w  = (const float*)d_in[27];
    const float* bm_in_w   = (const float*)d_in[28];
    const float* bm_conv_w = (const float*)d_in[29];
    const float* bm_conv_b = (const float*)d_in[30];
    const float* bm_xpw    = (const float*)d_in[31];
    const float* bm_dt_w   = (const float*)d_in[32];
    const float* bm_dt_b   = (const float*)d_in[33];
    const float* bm_A_log  = (const float*)d_in[34];
    const float* bm_D      = (const float*)d_in[35];
    const float* bm_out_w  = (const float*)d_in[36];
    float* out = (float*)d_out;

    // workspace carve-up (floats)
    float* ws = (float*)d_ws;
    size_t o  = 0;
    auto alloc = [&](size_t n) { float* p = ws + o; o += n; return p; };
    float* xn     = alloc((size_t)MR * DM);
    float* xnrev  = alloc((size_t)MR * DM);
    float* xz_f   = alloc((size_t)MR * 2 * DI_);
    float* xz_b   = alloc((size_t)MR * 2 * DI_);
    float* xc_f   = alloc((size_t)MR * DI_);
    float* xc_b   = alloc((size_t)MR * DI_);
    float* xdbl_f = alloc((size_t)MR * 64);
    float* xdbl_b = alloc((size_t)MR * 64);
    float* dlt_f  = alloc((size_t)MR * DI_);
    float* dlt_b  = alloc((size_t)MR * DI_);
    float* y_f    = alloc((size_t)MR * DI_);
    float* y_b    = alloc((size_t)MR * DI_);
    float* fbcat  = alloc((size_t)MR * 2 * DM);   // [f_out | unflipped b_out]
    float* m_out  = alloc((size_t)MR * DM);
    float* proj   = alloc((size_t)MR * DM);
    float* cc_out = alloc((size_t)MR * DM);
    float* x1     = alloc((size_t)MR * DM);
    float* hn     = alloc((size_t)MR * DM);
    float* ffn_h  = alloc((size_t)MR * 4 * DM);
    float* cnorm  = alloc(NCL);

    // 1) xn = LN(x); xnrev = flip(xn) along L
    ln2_kernel<<<MR, 256, 0, stream>>>(x, norm1_g, norm1_b, xn, xnrev);

    // 2) in-projections (TDM + WMMA): xz = xn @ in_w.T
    launch_gemm(0, xn,    DM, fm_in_w, DM, nullptr, nullptr, xz_f, 2 * DI_, 0, MR, 2 * DI_, DM, 0, stream);
    launch_gemm(0, xnrev, DM, bm_in_w, DM, nullptr, nullptr, xz_b, 2 * DI_, 0, MR, 2 * DI_, DM, 0, stream);

    // 3) depthwise causal conv + silu
    conv_silu_kernel<<<(MR * DI_) / 256, 256, 0, stream>>>(xz_f, fm_conv_w, fm_conv_b, xc_f);
    conv_silu_kernel<<<(MR * DI_) / 256, 256, 0, stream>>>(xz_b, bm_conv_w, bm_conv_b, xc_b);

    // 4) x_dbl = xc @ xproj_w.T  (dt|B|C)  -- N=64 -> v1 kernel
    launch_gemm(0, xc_f, DI_, fm_xpw, DI_, nullptr, nullptr, xdbl_f, 64, 0, MR, 64, DI_, 0, stream);
    launch_gemm(0, xc_b, DI_, bm_xpw, DI_, nullptr, nullptr, xdbl_b, 64, 0, MR, 64, DI_, 0, stream);

    // 5) delta = softplus(dt @ dt_w.T + dt_b)   (K=32 -> v1 kernel; A = first 32 cols of xdbl)
    launch_gemm(1, xdbl_f, 64, fm_dt_w, DTR, fm_dt_b, nullptr, dlt_f, DI_, 0, MR, DI_, DTR, 0, stream);
    launch_gemm(1, xdbl_b, 64, bm_dt_w, DTR, bm_dt_b, nullptr, dlt_b, DI_, 0, MR, DI_, DTR, 0, stream);

    // 6) selective scan (+ xc*D, * silu(z))
    ssm_scan_kernel<<<(BB * DI_) / 256, 256, 0, stream>>>(dlt_f, xc_f, xdbl_f, xz_f, fm_A_log, fm_D, y_f);
    ssm_scan_kernel<<<(BB * DI_) / 256, 256, 0, stream>>>(dlt_b, xc_b, xdbl_b, xz_b, bm_A_log, bm_D, y_b);

    // 7) out-projections into concat buffer; backward path un-flipped on store
    launch_gemm(0, y_f, DI_, fm_out_w, DI_, nullptr, nullptr, fbcat, 2 * DM, 0,  MR, DM, DI_, 0,  stream);
    launch_gemm(0, y_b, DI_, bm_out_w, DI_, nullptr, nullptr, fbcat, 2 * DM, DM, MR, DM, DI_, LL, stream);

    // 8) fusion: mamba_out = concat @ fusion_w.T + fusion_b
    launch_gemm(0, fbcat, 2 * DM, fusion_w, 2 * DM, fusion_b, nullptr, m_out, DM, 0, MR, DM, 2 * DM, 0, stream);

    // 9) context-clustering path
    launch_gemm(0, xn, DM, cc_pw, DM, cc_pb, nullptr, proj, DM, 0, MR, DM, DM, 0, stream);
    center_norm_kernel<<<NCL, 256, 0, stream>>>(cc_cent, cnorm);
    cc_kernel<<<MR, 256, 0, stream>>>(xn, proj, cc_cent, cnorm, cc_alpha, cc_ng, cc_nb, cc_out);

    // 10) gated fusion + residual: x1 = x + g0*mamba_out + g1*cc_out
    gate_fuse_kernel<<<MR, 256, 0, stream>>>(x, xn, gate_w, gate_b, m_out, cc_out, x1);

    // 11) FFN: out = x1 + gelu(LN(x1)@w1.T+b1)@w2.T+b2
    ln2_kernel<<<MR, 256, 0, stream>>>(x1, ffn_ng, ffn_nb, hn, nullptr);
    launch_gemm(2, hn,    DM,     ffn_w1, DM,     ffn_b1, nullptr, ffn_h, 4 * DM, 0, MR, 4 * DM, DM,     0, stream);
    launch_gemm(0, ffn_h, 4 * DM, ffn_w2, 4 * DM, ffn_b2, x1,      out,   DM,     0, MR, DM,     4 * DM, 0, stream);
}